// InfoNCELoss_55362128445929
// MI455X (gfx1250) — compile-verified
//
#include <hip/hip_runtime.h>
#include <math.h>

// InfoNCE on MI455X (gfx1250): HBM-bound streaming GEMV over the 256MB
// negative matrix. fp32 WMMA (V_WMMA_F32_16X16X4_F32) computes 16 row-dots
// per wave-tile; norms ride along on VALU from the same loaded registers.

typedef __attribute__((ext_vector_type(2))) float v2f;
typedef __attribute__((ext_vector_type(8))) float v8f;

#define INV_TEMP 20.0f
#define EPS_F 1e-6f

// ---------------- kernel 0: anchor/positive stats + zero accumulator -------
__global__ __launch_bounds__(256)
void infonce_stats_kernel(const float* __restrict__ a,
                          const float* __restrict__ p,
                          float* __restrict__ ws, int D) {
    __shared__ float red0[256], red1[256], red2[256];
    const int t = threadIdx.x;
    float saa = 0.f, sap = 0.f, spp = 0.f;
    for (int i = t; i < D; i += 256) {
        float x = a[i], y = p[i];
        saa = fmaf(x, x, saa);
        sap = fmaf(x, y, sap);
        spp = fmaf(y, y, spp);
    }
    red0[t] = saa; red1[t] = sap; red2[t] = spp;
    __syncthreads();
    for (int s = 128; s > 0; s >>= 1) {
        if (t < s) { red0[t] += red0[t + s]; red1[t] += red1[t + s]; red2[t] += red2[t + s]; }
        __syncthreads();
    }
    if (t == 0) {
        float na = sqrtf(red0[0]);
        float np = sqrtf(red2[0]);
        float pos_sim = red1[0] / fmaxf(na * np, EPS_F) * INV_TEMP;
        ws[0] = 0.f;      // exp-sum accumulator
        ws[1] = na;       // anchor norm
        ws[2] = pos_sim;  // scaled positive similarity
    }
}

// ---------------- kernel 1: stream negatives, WMMA dots + VALU norms -------
// grid: N/128 blocks, 256 threads (8 waves). Each wave owns a 16-row tile.
__global__ __launch_bounds__(256)
void infonce_neg_kernel(const float* __restrict__ neg,
                        const float* __restrict__ anchor,
                        float* __restrict__ ws, int D) {
    __shared__ float sA[1024];        // anchor staged once per block
    __shared__ float sDots[8][16];    // per-wave dot scratch

    const int t = threadIdx.x;
    for (int i = t; i < D; i += 256) sA[i] = anchor[i];
    __syncthreads();

    const int wave = t >> 5;
    const int lane = t & 31;
    const int m    = lane & 15;       // row within tile
    const bool hif = (lane >= 16);    // hi half-wave

    const long long tile = (long long)blockIdx.x * 8 + wave;
    const long long row  = tile * 16 + m;
    const float4* rowp = (const float4*)(neg + row * (long long)D);

    v8f c = {0.f, 0.f, 0.f, 0.f, 0.f, 0.f, 0.f, 0.f};
    float n2 = 0.f;
    const int koff = hif ? 1 : 0;     // float4-index offset within 8-col strip
    const int boff = hif ? 2 : 0;     // anchor K offset for B fragment

    for (int k0 = 0; k0 < D; k0 += 8) {
        // lo lanes: row m cols [k0,k0+4); hi lanes: row m cols [k0+4,k0+8)
        float4 g = rowp[(k0 >> 2) + koff];
        if (k0 + 64 < D)
            __builtin_prefetch((const void*)&rowp[((k0 + 64) >> 2) + koff], 0, 0);

        // per-row squared norm rides along on VALU
        n2 = fmaf(g.x, g.x, fmaf(g.y, g.y, fmaf(g.z, g.z, fmaf(g.w, g.w, n2))));

        // half-wave swap to build A fragments (A layout: lanes0-15 K=0,1; lanes16-31 K=2,3)
        float px = __shfl_xor(g.x, 16);
        float py = __shfl_xor(g.y, 16);
        float pz = __shfl_xor(g.z, 16);
        float pw = __shfl_xor(g.w, 16);

        v2f a0, a1, b0, b1;
        a0.x = hif ? pz  : g.x;  a0.y = hif ? pw  : g.y;   // K = k0 .. k0+3
        a1.x = hif ? g.z : px;   a1.y = hif ? g.w : py;    // K = k0+4 .. k0+7

        // B broadcast: every column of B holds anchor[k] -> all 16 C columns
        // accumulate the identical 16 row-dots.
        b0.x = sA[k0 + boff];     b0.y = sA[k0 + boff + 1];
        b1.x = sA[k0 + 4 + boff]; b1.y = sA[k0 + 4 + boff + 1];

        c = __builtin_amdgcn_wmma_f32_16x16x4_f32(false, a0, false, b0,
                                                  (short)0, c, false, false);
        c = __builtin_amdgcn_wmma_f32_16x16x4_f32(false, a1, false, b1,
                                                  (short)0, c, false, false);
    }

    // combine the two half-row norm partials: lane m <- + lane m+16
    n2 += __shfl_xor(n2, 16);

    // C layout: lanes 0-15 hold rows 0..7 in c[0..7] (any column),
    //           lanes 16-31 hold rows 8..15. Extract via LDS.
    if (lane == 0) {
        for (int r = 0; r < 8; ++r) sDots[wave][r] = c[r];
    }
    if (lane == 16) {
        for (int r = 0; r < 8; ++r) sDots[wave][8 + r] = c[r];
    }
    __syncthreads();

    float e = 0.f;
    if (!hif) {
        float na   = ws[1];
        float dot  = sDots[wave][m];
        float cosv = dot / fmaxf(na * sqrtf(n2), EPS_F);
        e = expf(cosv * INV_TEMP);
    }
    // wave-level sum (hi lanes contribute 0)
    for (int o = 16; o > 0; o >>= 1) e += __shfl_xor(e, o);
    if (lane == 0) atomicAdd(&ws[0], e);
}

// ---------------- kernel 2: finisher ---------------------------------------
__global__ void infonce_finish_kernel(const float* __restrict__ ws,
                                      float* __restrict__ out) {
    float neg_sum = ws[0];
    float pos_sim = ws[2];
    out[0] = -logf(expf(pos_sim) / neg_sum + EPS_F);
}

extern "C" void kernel_launch(void* const* d_in, const int* in_sizes, int n_in,
                              void* d_out, int out_size, void* d_ws, size_t ws_size,
                              hipStream_t stream) {
    const float* anchor   = (const float*)d_in[0];
    const float* positive = (const float*)d_in[1];
    const float* negative = (const float*)d_in[2];
    float* out = (float*)d_out;
    float* ws  = (float*)d_ws;

    const int D = in_sizes[0];            // 1024
    const int N = in_sizes[2] / D;        // 65536

    infonce_stats_kernel<<<1, 256, 0, stream>>>(anchor, positive, ws, D);

    // 8 waves/block, 16 rows/wave -> 128 rows per block
    const int blocks = N / 128;           // 512
    infonce_neg_kernel<<<blocks, 256, 0, stream>>>(negative, anchor, ws, D);

    infonce_finish_kernel<<<1, 1, 0, stream>>>(ws, out);
}